// OTPE_63273458205170
// MI455X (gfx1250) — compile-verified
//
#include <hip/hip_runtime.h>
#include <math.h>

// OTPE timestep for MI455X (gfx1250, wave32, WMMA).
// Inputs (d_in order): x[8192], W[8192*4096], u[4096], E[8192*4096],
//                      R_hat[8192*4096], g_bar[4096], ratio[1]
// Output (concat):     s[4096] | u_new[4096] | E_new[33554432] |
//                      R_new[33554432] | g_bar_new[4096] | ratio_new[1]

#define N_IN   8192
#define N_OUT  4096
#define SIG_TAU 0.8807970779778823f   // sigmoid(2.0)
#define V_TH    1.0f
#define SLOPE   10.0f
#define KSPLIT  16
#define KCHUNK  (N_IN / KSPLIT)       // 512

typedef __attribute__((ext_vector_type(2))) float v2f;
typedef __attribute__((ext_vector_type(8))) float v8f;

// ---------------------------------------------------------------------------
// Kernel 1: GEMV partials via V_WMMA_F32_16X16X4_F32.
// One wave per block. Each wave owns 32 output columns (two 16-col WMMA tiles,
// giving a full 128B cacheline per W row) and a 512-deep K slice.
// A-matrix (16x4 f32): lanes 0-15 hold M=lane, lanes 16-31 hold M=lane-16;
// VGPR0 = K=0 (lo half) / K=2 (hi half), VGPR1 = K=1 / K=3. Only row M=0
// carries x; all other rows are zero, so D row 0 = x-chunk dot W-tile.
// B-matrix (4x16 f32): VGPR0 = row K=0 (lanes 0-15) / K=2 (lanes 16-31),
// VGPR1 = row K=1 / K=3, N = lane&15.
// ---------------------------------------------------------------------------
__global__ __launch_bounds__(32) void otpe_gemv_wmma(
    const float* __restrict__ x, const float* __restrict__ W,
    float* __restrict__ part /* [KSPLIT][N_OUT] */) {
  const int lane = threadIdx.x;
  const int half = lane >> 4;     // 0: K=0,1   1: K=2,3
  const int l15  = lane & 15;
  const int j0   = blockIdx.x * 32;
  const int k0   = blockIdx.y * KCHUNK;

  v8f c0 = {};
  v8f c1 = {};

#pragma unroll 2
  for (int k = k0; k < k0 + KCHUNK; k += 4) {
    const int kr = k + 2 * half;          // this lane's K rows: kr, kr+1
    v2f a;
    a[0] = (l15 == 0) ? x[kr]     : 0.0f; // A row M=0 only
    a[1] = (l15 == 0) ? x[kr + 1] : 0.0f;

    const float* wp = W + (size_t)kr * N_OUT + j0 + l15;
    v2f b0, b1;
    b0[0] = wp[0];          // tile 0: row kr
    b0[1] = wp[N_OUT];      //         row kr+1
    b1[0] = wp[16];         // tile 1
    b1[1] = wp[N_OUT + 16];

    c0 = __builtin_amdgcn_wmma_f32_16x16x4_f32(false, a, false, b0,
                                               (short)0, c0, false, false);
    c1 = __builtin_amdgcn_wmma_f32_16x16x4_f32(false, a, false, b1,
                                               (short)0, c1, false, false);
  }

  // D row M=0 lives in accumulator VGPR0, lanes 0..15 (N = lane).
  if (half == 0) {
    float* p = part + (size_t)blockIdx.y * N_OUT + j0 + l15;
    p[0]  = c0[0];
    p[16] = c1[0];
  }
}

// ---------------------------------------------------------------------------
// Kernel 2: deterministic fixed-order reduction of K-split partials + all
// small per-column outputs.  ds_du_prev / sig_tau == sg.
// ---------------------------------------------------------------------------
__global__ __launch_bounds__(256) void otpe_pointwise(
    const float* __restrict__ u, const float* __restrict__ g_bar,
    const float* __restrict__ ratio, const float* __restrict__ part,
    float* __restrict__ sg_ws,
    float* __restrict__ out_s, float* __restrict__ out_u,
    float* __restrict__ out_gbar, float* __restrict__ out_ratio) {
  const int j = blockIdx.x * blockDim.x + threadIdx.x;
  if (j >= N_OUT) return;

  float acc = SIG_TAU * u[j];
  for (int p = 0; p < KSPLIT; ++p) acc += part[(size_t)p * N_OUT + j];

  const float s  = (acc >= V_TH) ? 1.0f : 0.0f;
  const float t  = 1.0f + SLOPE * fabsf(acc - V_TH);
  const float sg = 1.0f / (t * t);

  out_s[j]  = s;
  out_u[j]  = acc - s * V_TH;
  sg_ws[j]  = sg;

  const float r0 = SIG_TAU * ratio[0];
  const float rn = r0 + 1.0f;
  const float r  = r0 / rn;
  out_gbar[j] = r * g_bar[j] + (1.0f - r) * sg;   // (ds_du_prev/sig_tau)==sg
  if (j == 0) out_ratio[0] = rn;
}

// ---------------------------------------------------------------------------
// Kernel 3: bandwidth-bound trace update, float4 vectorized.
//   E_new = sig_tau*E + x[i]
//   R_new = sig_tau*R_hat + sg[j]*(sig_tau*E + x[i]) = sig_tau*R_hat + sg*E_new
// ---------------------------------------------------------------------------
__global__ __launch_bounds__(256) void otpe_trace(
    const float* __restrict__ x, const float* __restrict__ E,
    const float* __restrict__ Rh, const float* __restrict__ sg,
    float* __restrict__ E_new, float* __restrict__ R_new) {
  const size_t t4   = (size_t)blockIdx.x * blockDim.x + threadIdx.x;
  const size_t base = t4 * 4;               // element index; row len 4096 % 4 == 0
  const int i = (int)(base >> 12);          // / N_OUT
  const int j = (int)(base & (N_OUT - 1));

  const float  xi = x[i];
  const float4 e  = *(const float4*)(E  + base);
  const float4 rh = *(const float4*)(Rh + base);
  const float4 g  = *(const float4*)(sg + j);

  float4 en, rn;
  en.x = fmaf(SIG_TAU, e.x, xi);
  en.y = fmaf(SIG_TAU, e.y, xi);
  en.z = fmaf(SIG_TAU, e.z, xi);
  en.w = fmaf(SIG_TAU, e.w, xi);
  rn.x = fmaf(g.x, en.x, SIG_TAU * rh.x);
  rn.y = fmaf(g.y, en.y, SIG_TAU * rh.y);
  rn.z = fmaf(g.z, en.z, SIG_TAU * rh.z);
  rn.w = fmaf(g.w, en.w, SIG_TAU * rh.w);

  *(float4*)(E_new + base) = en;
  *(float4*)(R_new + base) = rn;
}

// ---------------------------------------------------------------------------
extern "C" void kernel_launch(void* const* d_in, const int* in_sizes, int n_in,
                              void* d_out, int out_size, void* d_ws, size_t ws_size,
                              hipStream_t stream) {
  (void)in_sizes; (void)n_in; (void)out_size; (void)ws_size;

  const float* x     = (const float*)d_in[0];
  const float* W     = (const float*)d_in[1];
  const float* u     = (const float*)d_in[2];
  const float* E     = (const float*)d_in[3];
  const float* R_hat = (const float*)d_in[4];
  const float* g_bar = (const float*)d_in[5];
  const float* ratio = (const float*)d_in[6];

  float* out = (float*)d_out;
  const size_t MAT = (size_t)N_IN * N_OUT;      // 33554432
  float* out_s     = out;
  float* out_u     = out + N_OUT;
  float* out_E     = out + 2 * (size_t)N_OUT;
  float* out_R     = out_E + MAT;
  float* out_gbar  = out_R + MAT;
  float* out_ratio = out_gbar + N_OUT;

  float* part  = (float*)d_ws;                  // [KSPLIT][N_OUT]
  float* sg_ws = part + (size_t)KSPLIT * N_OUT; // [N_OUT]

  // 1) GEMV partials via WMMA: 128 col-tiles x 16 K-splits, 1 wave each.
  otpe_gemv_wmma<<<dim3(N_OUT / 32, KSPLIT), 32, 0, stream>>>(x, W, part);

  // 2) Per-column pointwise outputs (deterministic reduction of partials).
  otpe_pointwise<<<N_OUT / 256, 256, 0, stream>>>(
      u, g_bar, ratio, part, sg_ws, out_s, out_u, out_gbar, out_ratio);

  // 3) Trace matrices: 33.5M elements / 4 per thread / 256 per block.
  const int nblk = (int)(MAT / 4 / 256);        // 32768
  otpe_trace<<<nblk, 256, 0, stream>>>(x, E, R_hat, sg_ws, out_E, out_R);
}